// KNNLayer_39444979647064
// MI455X (gfx1250) — compile-verified
//
#include <hip/hip_runtime.h>

// gfx1250 / MI455X fused KNN: WMMA f32 16x16x4 distance tiles + in-register top-k.

typedef float v2f __attribute__((ext_vector_type(2)));
typedef float v8f __attribute__((ext_vector_type(8)));

#define BATCH 4
#define NPTS  8192
#define DIM   16
#define KSEL  16
#define ROWS  16           // query rows per block
#define SCOLS 256          // candidate columns per super-tile
#define NWAVES 8
#define TPW   2            // 16x16 tiles per wave per super-tile (256/16/8)
#define LPAD  (SCOLS + 4)  // LDS row padding to stagger banks

__global__ __launch_bounds__(256, 1)
void knn_topk_kernel(const float* __restrict__ X, int* __restrict__ out)
{
    __shared__ float distT[ROWS][LPAD];
    __shared__ float candD[ROWS][LPAD];
    __shared__ int   candI[ROWS][LPAD];

    const int tid  = threadIdx.x;
    const int lane = tid & 31;
    const int wave = tid >> 5;
    const int hi   = lane >> 4;       // which half of the wave
    const int ln16 = lane & 15;

    const int b  = blockIdx.x >> 9;            // 512 row-tiles per batch
    const int m0 = (blockIdx.x & 511) << 4;    // base query row

    const float* Xb = X + (size_t)b * NPTS * DIM;

    // ---- A tile (loop-invariant): 16 query rows, f32 16x4 WMMA layout ----
    // lane holds row m = ln16; VGPR pair per K-chunk kk covers K = 4*kk + 2*hi + {0,1}
    v2f amat[4];
    float sqa_part = 0.0f;
    {
        const float* rowp = Xb + (size_t)(m0 + ln16) * DIM + hi * 2;
        #pragma unroll
        for (int kk = 0; kk < 4; ++kk) {
            v2f v = *(const v2f*)(rowp + kk * 4);
            amat[kk] = v;
            sqa_part += v.x * v.x + v.y * v.y;
        }
    }
    // full ||x_m||^2 : combine the two half-wave partials for the same row
    float sqa = sqa_part + __shfl_xor(sqa_part, 16, 32);
    // per-lane copies for the 8 C rows this lane covers (M = r + 8*hi)
    float sqAm[8];
    #pragma unroll
    for (int r = 0; r < 8; ++r)
        sqAm[r] = __shfl(sqa, r + 8 * hi, 32);

    // ---- per-thread sorted top-k list (registers only) ----
    float bd[KSEL];
    int   bi[KSEL];
    #pragma unroll
    for (int j = 0; j < KSEL; ++j) { bd[j] = 3.4e38f; bi[j] = 0; }

    const int myrow = tid >> 4;   // consumer row 0..15
    const int mycol = tid & 15;   // strided column slot

    for (int s0 = 0; s0 < NPTS; s0 += SCOLS) {
        // ============ produce: WMMA distance tiles into LDS ============
        #pragma unroll
        for (int tt = 0; tt < TPW; ++tt) {
            const int tileCol = (wave * TPW + tt) * 16;    // within super-tile
            const int n       = s0 + tileCol + ln16;       // global column
            const float* colp = Xb + (size_t)n * DIM + hi * 2;

            v2f bmat[4];
            float sqb_part = 0.0f;
            #pragma unroll
            for (int kk = 0; kk < 4; ++kk) {
                v2f v = *(const v2f*)(colp + kk * 4);
                bmat[kk] = v;
                sqb_part += v.x * v.x + v.y * v.y;
            }
            float sqb = sqb_part + __shfl_xor(sqb_part, 16, 32);

            // inner[m][n] for the 16x16 tile: K=16 as 4 chained k=4 WMMAs
            v8f c = {};
            #pragma unroll
            for (int kk = 0; kk < 4; ++kk)
                c = __builtin_amdgcn_wmma_f32_16x16x4_f32(
                        false, amat[kk], false, bmat[kk],
                        (short)0, c, false, false);

            // dist = ||x_m||^2 + ||x_n||^2 - 2*inner  (C row M = r + 8*hi, col N = ln16)
            #pragma unroll
            for (int r = 0; r < 8; ++r) {
                float d = sqAm[r] + sqb - 2.0f * c[r];
                distT[r + 8 * hi][tileCol + ln16] = d;
            }
        }
        __syncthreads();

        // ============ consume: strided insertion into top-k ============
        #pragma unroll
        for (int j = 0; j < 16; ++j) {
            const int cc = mycol + j * 16;
            float d  = distT[myrow][cc];
            int   id = s0 + cc;
            if (d < bd[KSEL - 1]) {            // fast-reject common case
                bd[KSEL - 1] = d;
                bi[KSEL - 1] = id;
                #pragma unroll
                for (int q = KSEL - 1; q > 0; --q) {
                    if (bd[q] < bd[q - 1]) {
                        float td = bd[q]; bd[q] = bd[q - 1]; bd[q - 1] = td;
                        int   ti = bi[q]; bi[q] = bi[q - 1]; bi[q - 1] = ti;
                    }
                }
            }
        }
        __syncthreads();
    }

    // ---- dump 16 partial lists per row (256 candidates/row) ----
    #pragma unroll
    for (int j = 0; j < KSEL; ++j) {
        candD[myrow][mycol * 16 + j] = bd[j];
        candI[myrow][mycol * 16 + j] = bi[j];
    }
    __syncthreads();

    // ---- final selection: one thread per row, 16 argmin passes ----
    if (tid < ROWS) {
        const int row = tid;
        int* orow = out + ((size_t)b * NPTS + (m0 + row)) * KSEL;
        for (int kk = 0; kk < KSEL; ++kk) {
            float best = 3.4e38f;
            int bestj = 0;
            for (int j = 0; j < SCOLS; ++j) {
                float d = candD[row][j];
                if (d < best) { best = d; bestj = j; }
            }
            orow[kk] = candI[row][bestj];
            candD[row][bestj] = 3.4e38f;   // exclude and continue
        }
    }
}

extern "C" void kernel_launch(void* const* d_in, const int* in_sizes, int n_in,
                              void* d_out, int out_size, void* d_ws, size_t ws_size,
                              hipStream_t stream) {
    (void)in_sizes; (void)n_in; (void)d_ws; (void)ws_size; (void)out_size;
    const float* X = (const float*)d_in[0];   // [4, 8192, 16] f32
    int* out = (int*)d_out;                   // [4, 8192, 16] int32 indices
    const int nblocks = BATCH * (NPTS / ROWS);      // 4 * 512 = 2048
    knn_topk_kernel<<<nblocks, 256, 0, stream>>>(X, out);
}